// QuantumMultiHeadAttention_91199335563770
// MI455X (gfx1250) — compile-verified
//
#include <hip/hip_runtime.h>
#include <hip/hip_bf16.h>

// ---------------------------------------------------------------------------
// Quantum multi-head attention, fused for gfx1250 (MI455X), f32 WMMA path.
//   B=4, S=512, D_MODEL=64, H=4, NQ=4, d_k=16, CRY theta = pi/4.
// Core trick: Re<q|k> and Im<q|k> are dot4() of real 4-vectors, so the S x S
// overlap matrices per (head,qubit) are rank-4 GEMMs -> V_WMMA_F32_16X16X4_F32.
// ---------------------------------------------------------------------------

#define BB 4
#define SS 512
#define DM 64
#define NH 4
#define NQ 4

typedef float v2f __attribute__((ext_vector_type(2)));
typedef float v8f __attribute__((ext_vector_type(8)));

#define CRY_C 0.70710678118654752f  // cos(pi/4)
#define CRY_S 0.70710678118654752f  // sin(pi/4)
#define PI_F  3.14159265358979323846f

// ---------------------------------------------------------------------------
// psi = RZ(omega) RY(theta) RZ(phi) RY(data*pi) |0>
// ---------------------------------------------------------------------------
__device__ __forceinline__ void qubit_state(float data, float phi, float theta,
                                            float omega, float& u0r, float& u0i,
                                            float& u1r, float& u1i) {
    float a  = data * PI_F;
    float c0 = cosf(0.5f * a), s0 = sinf(0.5f * a);
    float cp = cosf(0.5f * phi), sp = sinf(0.5f * phi);
    // v0 = c0 * exp(-i phi/2), v1 = s0 * exp(+i phi/2)
    float v0r = c0 * cp, v0i = -c0 * sp;
    float v1r = s0 * cp, v1i =  s0 * sp;
    float ct = cosf(0.5f * theta), st = sinf(0.5f * theta);
    float w0r = ct * v0r - st * v1r, w0i = ct * v0i - st * v1i;
    float w1r = st * v0r + ct * v1r, w1i = st * v0i + ct * v1i;
    float co = cosf(0.5f * omega), so = sinf(0.5f * omega);
    // u0 *= exp(-i omega/2); u1 *= exp(+i omega/2)
    u0r = w0r * co + w0i * so;  u0i = w0i * co - w0r * so;
    u1r = w1r * co - w1i * so;  u1i = w1i * co + w1r * so;
}

// ---------------------------------------------------------------------------
// Stage 1: per (b,s) compute state vectors for q/k/v over all (h,n).
//   qR[idx] = ( q0r,  q0i,  q1r, q1i)   -> A-frag for Re<q|k>
//   qI[idx] = (-q0i,  q0r, -q1i, q1r)   -> A-frag for Im<q|k>
//   kv[idx] = ( k0r,  k0i,  k1r, k1i)   -> B-frag
//   zArr/xArr: per (b,s,h) float4 over n of <Z>, <X> of psi_v.
// ---------------------------------------------------------------------------
__global__ void k1_prep(const float* __restrict__ q, const float* __restrict__ k,
                        const float* __restrict__ v, const float* __restrict__ Wq,
                        const float* __restrict__ Wk, const float* __restrict__ Wv,
                        float4* __restrict__ qR, float4* __restrict__ qI,
                        float4* __restrict__ kv, float4* __restrict__ zArr,
                        float4* __restrict__ xArr) {
    int t = blockIdx.x * blockDim.x + threadIdx.x;
    if (t >= BB * SS) return;
    #pragma unroll
    for (int h = 0; h < NH; ++h) {
        float4 zv, xv;
        float* zp = &zv.x;
        float* xp = &xv.x;
        #pragma unroll
        for (int n = 0; n < NQ; ++n) {
            int wb  = (h * NQ + n) * 3;
            int idx = (t * NH + h) * NQ + n;
            float u0r, u0i, u1r, u1i;
            // Q
            float dq = tanhf(q[t * DM + h * 16 + n]);
            qubit_state(dq, Wq[wb], Wq[wb + 1], Wq[wb + 2], u0r, u0i, u1r, u1i);
            qR[idx] = make_float4(u0r, u0i, u1r, u1i);
            qI[idx] = make_float4(-u0i, u0r, -u1i, u1r);
            // K
            float dk = tanhf(k[t * DM + h * 16 + n]);
            qubit_state(dk, Wk[wb], Wk[wb + 1], Wk[wb + 2], u0r, u0i, u1r, u1i);
            kv[idx] = make_float4(u0r, u0i, u1r, u1i);
            // V -> z, x
            float dv = tanhf(v[t * DM + h * 16 + n]);
            qubit_state(dv, Wv[wb], Wv[wb + 1], Wv[wb + 2], u0r, u0i, u1r, u1i);
            zp[n] = (u0r * u0r + u0i * u0i) - (u1r * u1r + u1i * u1i);
            xp[n] = 2.0f * (u0r * u1r + u0i * u1i);
        }
        zArr[t * NH + h] = zv;
        xArr[t * NH + h] = xv;
    }
}

// ---------------------------------------------------------------------------
// Stage 2: one wave per (b, i-block of 16, h). Sweep all 32 j-tiles.
// Per tile, per qubit: Re/Im overlap tiles via V_WMMA_F32_16X16X4_F32.
// A/B f32 fragments (ISA 7.12.2): lanes 0-15 hold K={0,1} in v[0],v[1];
// lanes 16-31 hold K={2,3} -> one float2 load at base + idx*16 + half*8.
// D layout: d[r] = element (row r + 8*half, col lane&15).
// ---------------------------------------------------------------------------
__global__ void __launch_bounds__(32) k2_pairs(
    const float4* __restrict__ qR, const float4* __restrict__ qI,
    const float4* __restrict__ kv, const float4* __restrict__ zArr,
    const float4* __restrict__ xArr, float* __restrict__ attended,
    float* __restrict__ attn) {
    const int lane   = threadIdx.x;
    const int half   = lane >> 4;
    const int lane16 = lane & 15;
    const int i0 = blockIdx.x * 16;  // i-block
    const int h  = blockIdx.y;
    const int b  = blockIdx.z;

    // Hoist A fragments (depend only on the i-block) out of the j loop.
    v2f aR[NQ], aI[NQ];
    #pragma unroll
    for (int n = 0; n < NQ; ++n) {
        int idx = (((b * SS + i0 + lane16) * NH + h) * NQ + n);
        aR[n] = *(const v2f*)(((const float*)qR) + idx * 4 + half * 2);
        aI[n] = *(const v2f*)(((const float*)qI) + idx * 4 + half * 2);
    }

    float acc[8][NQ];
    #pragma unroll
    for (int r = 0; r < 8; ++r)
        #pragma unroll
        for (int n = 0; n < NQ; ++n) acc[r][n] = 0.0f;

    for (int jt = 0; jt < SS / 16; ++jt) {
        const int j = jt * 16 + lane16;
        // F = prod_n |<q_n|k_n>|^2 over the 16x16 pair tile
        v8f F;
        #pragma unroll
        for (int n = 0; n < NQ; ++n) {
            int kidx = (((b * SS + j) * NH + h) * NQ + n);
            v2f bf = *(const v2f*)(((const float*)kv) + kidx * 4 + half * 2);
            v8f z8 = {};
            v8f re = __builtin_amdgcn_wmma_f32_16x16x4_f32(
                false, aR[n], false, bf, (short)0, z8, false, false);
            v8f im = __builtin_amdgcn_wmma_f32_16x16x4_f32(
                false, aI[n], false, bf, (short)0, z8, false, false);
            v8f m = re * re + im * im;
            F = (n == 0) ? m : F * m;
        }
        // p1 = 0.5(1-F); coef_z = 1 - (1-c)p1; coef_x = -s p1
        v8f p1 = 0.5f * (1.0f - F);
        v8f cz = 1.0f - (1.0f - CRY_C) * p1;
        v8f cx = (-CRY_S) * p1;

        const int zidx = (b * SS + j) * NH + h;
        float4 zj = zArr[zidx];
        float4 xj = xArr[zidx];
        const float* zp = &zj.x;
        const float* xp = &xj.x;

        #pragma unroll
        for (int r = 0; r < 8; ++r) {
            float czr = cz[r], cxr = cx[r];
            #pragma unroll
            for (int n = 0; n < NQ; ++n)
                acc[r][n] += czr * zp[n] + cxr * xp[n];
        }
        if (h == 0) {  // uniform per wave
            #pragma unroll
            for (int r = 0; r < 8; ++r) {
                int i = i0 + r + half * 8;
                attn[(b * SS + i) * SS + j] = cz[r] * zj.x + cx[r] * xj.x;
            }
        }
    }

    // Reduce the 16 columns held per lane-half (wave32 shuffle).
    #pragma unroll
    for (int r = 0; r < 8; ++r)
        #pragma unroll
        for (int n = 0; n < NQ; ++n) {
            float s = acc[r][n];
            s += __shfl_xor(s, 1);
            s += __shfl_xor(s, 2);
            s += __shfl_xor(s, 4);
            s += __shfl_xor(s, 8);
            acc[r][n] = s;
        }
    if (lane16 == 0) {
        #pragma unroll
        for (int r = 0; r < 8; ++r) {
            int i = i0 + r + half * 8;
            #pragma unroll
            for (int n = 0; n < NQ; ++n)
                attended[((b * SS + i) * NH + h) * NQ + n] =
                    acc[r][n] * (1.0f / (float)SS);
        }
    }
}

// ---------------------------------------------------------------------------
// Stage 3: out[row, d] = sum_{m<16} attended[row, m] * proj_w[d, m] + proj_b[d]
// (zero-padded cols 16..63 contribute nothing). [2048,16]x[16,64] via 4 chained
// K=4 WMMAs per 16x16 output tile.
// ---------------------------------------------------------------------------
__global__ void __launch_bounds__(32) k3_proj(const float* __restrict__ attended,
                                              const float* __restrict__ pw,
                                              const float* __restrict__ pb,
                                              float* __restrict__ out) {
    const int lane   = threadIdx.x;
    const int half   = lane >> 4;
    const int lane16 = lane & 15;
    const int rt = blockIdx.x;  // 0..127 row tiles of 2048
    const int ct = blockIdx.y;  // 0..3  col tiles of 64
    const int row = rt * 16 + lane16;
    const int d   = ct * 16 + lane16;

    v8f acc = {};
    #pragma unroll
    for (int kb = 0; kb < 16; kb += 4) {
        v2f a = *(const v2f*)(attended + row * 16 + kb + half * 2);
        v2f bf = *(const v2f*)(pw + d * DM + kb + half * 2);  // B[k][d] = pw[d,k]
        acc = __builtin_amdgcn_wmma_f32_16x16x4_f32(
            false, a, false, bf, (short)0, acc, false, false);
    }
    float bias = pb[d];
    #pragma unroll
    for (int r = 0; r < 8; ++r) {
        int rg = rt * 16 + r + half * 8;
        out[rg * DM + d] = acc[r] + bias;
    }
}

// ---------------------------------------------------------------------------
extern "C" void kernel_launch(void* const* d_in, const int* in_sizes, int n_in,
                              void* d_out, int out_size, void* d_ws, size_t ws_size,
                              hipStream_t stream) {
    const float* q  = (const float*)d_in[0];
    const float* k  = (const float*)d_in[1];
    const float* v  = (const float*)d_in[2];
    const float* Wq = (const float*)d_in[3];
    const float* Wk = (const float*)d_in[4];
    const float* Wv = (const float*)d_in[5];
    const float* pw = (const float*)d_in[6];
    const float* pb = (const float*)d_in[7];

    float* out  = (float*)d_out;                  // [4,512,64]  = 131072 f32
    float* attn = out + BB * SS * DM;             // [4,512,512] = 1048576 f32

    const int NSTATE = BB * SS * NH * NQ;         // 32768
    float4* qR   = (float4*)d_ws;
    float4* qI   = qR + NSTATE;
    float4* kv   = qI + NSTATE;
    float4* zArr = kv + NSTATE;                   // BB*SS*NH = 8192 float4
    float4* xArr = zArr + BB * SS * NH;
    float* attended = (float*)(xArr + BB * SS * NH);  // BB*SS*16 f32

    k1_prep<<<(BB * SS + 255) / 256, 256, 0, stream>>>(q, k, v, Wq, Wk, Wv, qR,
                                                       qI, kv, zArr, xArr);
    k2_pairs<<<dim3(SS / 16, NH, BB), 32, 0, stream>>>(qR, qI, kv, zArr, xArr,
                                                       attended, attn);
    k3_proj<<<dim3((BB * SS) / 16, DM / 16), 32, 0, stream>>>(attended, pw, pb,
                                                              out);
}